// GraphAwareAttention_13572096655642
// MI455X (gfx1250) — compile-verified
//
#include <hip/hip_runtime.h>

#define NN 4096
#define DD 256
#define HH 8
#define DKK 32
#define EE 131072

typedef __attribute__((ext_vector_type(2))) float v2f;
typedef __attribute__((ext_vector_type(8))) float v8f;

// D = A(16x4 f32) * B(4x16 f32) + C(16x16 f32), full fp32 precision.
__device__ __forceinline__ v8f wmma_f32(v2f a, v2f b, v8f c) {
  return __builtin_amdgcn_wmma_f32_16x16x4_f32(
      /*neg_a=*/false, a, /*neg_b=*/false, b,
      /*c_mod=*/(short)0, c, /*reuse_a=*/false, /*reuse_b=*/false);
}

// ---------------------------------------------------------------------------
// GEMM: Y[rows, 256] = X[rows, 256] @ W^T[256, 256] + bias   (nn.Linear)
// One wave computes a 16x64 output strip (A regs reused across 4 B tiles).
// grid = (rows/16, 256/64), block = 32 (wave32).
// ---------------------------------------------------------------------------
__global__ void __launch_bounds__(32)
gemm_xwT_kernel(const float* __restrict__ X, const float* __restrict__ W,
                const float* __restrict__ bias, float* __restrict__ Y) {
  const int r0 = blockIdx.x * 16;
  const int c0 = blockIdx.y * 64;
  const int lane = threadIdx.x;
  const int n = lane & 15;       // row (A) / col (B,C) within tile
  const int half = lane >> 4;    // K-split for A/B operands

  v8f acc0 = 0.f, acc1 = 0.f, acc2 = 0.f, acc3 = 0.f;
  for (int k = 0; k < DD; k += 4) {
    v2f a = *(const v2f*)&X[(size_t)(r0 + n) * DD + k + 2 * half];
    v2f b0 = *(const v2f*)&W[(size_t)(c0 + 0 * 16 + n) * DD + k + 2 * half];
    v2f b1 = *(const v2f*)&W[(size_t)(c0 + 1 * 16 + n) * DD + k + 2 * half];
    v2f b2 = *(const v2f*)&W[(size_t)(c0 + 2 * 16 + n) * DD + k + 2 * half];
    v2f b3 = *(const v2f*)&W[(size_t)(c0 + 3 * 16 + n) * DD + k + 2 * half];
    acc0 = wmma_f32(a, b0, acc0);
    acc1 = wmma_f32(a, b1, acc1);
    acc2 = wmma_f32(a, b2, acc2);
    acc3 = wmma_f32(a, b3, acc3);
  }
  v8f* accs[4] = {&acc0, &acc1, &acc2, &acc3};
#pragma unroll
  for (int c = 0; c < 4; ++c) {
    float bcol = bias[c0 + c * 16 + n];
#pragma unroll
    for (int v = 0; v < 8; ++v) {
      int row = r0 + v + 8 * half;  // C layout: M = vgpr + 8*(lane>=16)
      Y[(size_t)row * DD + c0 + c * 16 + n] = (*accs[c])[v] + bcol;
    }
  }
}

// ---------------------------------------------------------------------------
// Dense edge-weight matrix: sentinel init + scatter (non-accumulating set).
// ---------------------------------------------------------------------------
__global__ void __launch_bounds__(256)
init_dense_kernel(float4* __restrict__ Wd4) {
  size_t i = (size_t)blockIdx.x * 256 + threadIdx.x;  // NN*NN/4 elements
  Wd4[i] = make_float4(-1.f, -1.f, -1.f, -1.f);
}

__global__ void __launch_bounds__(256)
scatter_kernel(const long long* __restrict__ ei, const float* __restrict__ ew,
               float* __restrict__ Wd) {
  int e = blockIdx.x * 256 + threadIdx.x;
  if (e < EE) {
    int s = (int)ei[e];
    int d = (int)ei[EE + e];
    Wd[(size_t)s * NN + d] = ew[e];
  }
}

// ---------------------------------------------------------------------------
// Fused flash attention with dense edge bias.
// grid = (NN/16, HH), block = 32. One wave owns 16 query rows of one head.
// ---------------------------------------------------------------------------
__global__ void __launch_bounds__(32)
attn_kernel(const float* __restrict__ Q, const float* __restrict__ K,
            const float* __restrict__ V, const float* __restrict__ Wd,
            const float* __restrict__ we, const float* __restrict__ be,
            float* __restrict__ ctx) {
  __shared__ float sP[16][16];
  const int q0 = blockIdx.x * 16;
  const int h = blockIdx.y;
  const int lane = threadIdx.x;
  const int n = lane & 15;
  const int half = lane >> 4;
  const float scale = 0.17677669529663687f;  // 1/sqrt(DK=32)
  const float weh = we[h];
  const float beh = be[h];

  // Preload Q tile in A-layout: 8 slices of 16x4 covering DK=32.
  v2f qa[8];
#pragma unroll
  for (int s = 0; s < 8; ++s)
    qa[s] = *(const v2f*)&Q[(size_t)(q0 + n) * DD + h * DKK + 4 * s + 2 * half];

  v8f o0 = 0.f, o1 = 0.f;           // 16x32 output accumulator (2 col tiles)
  float mrow[8], lrow[8];           // per-row online-softmax stats
#pragma unroll
  for (int v = 0; v < 8; ++v) { mrow[v] = -3.0e38f; lrow[v] = 0.f; }

  for (int k0 = 0; k0 < NN; k0 += 16) {
    // ---- S = (Q @ K^T) for this 16x16 tile (8 chained f32 WMMAs) ----
    v8f S = 0.f;
#pragma unroll
    for (int s = 0; s < 8; ++s) {
      v2f bk = *(const v2f*)&K[(size_t)(k0 + n) * DD + h * DKK + 4 * s + 2 * half];
      S = wmma_f32(qa[s], bk, S);
    }
    // ---- scale + edge bias (dense lookup, sentinel < 0 == no edge) ----
#pragma unroll
    for (int v = 0; v < 8; ++v) {
      float sv = S[v] * scale;
      float w = Wd[(size_t)(q0 + v + 8 * half) * NN + (k0 + n)];
      if (w >= 0.f) sv += weh * w + beh;
      S[v] = sv;
    }
    // ---- online softmax (row reductions across each 16-lane half) ----
#pragma unroll
    for (int v = 0; v < 8; ++v) {
      float rmax = S[v];
#pragma unroll
      for (int off = 8; off >= 1; off >>= 1)
        rmax = fmaxf(rmax, __shfl_xor(rmax, off, 32));
      float mnew = fmaxf(mrow[v], rmax);
      float corr = __expf(mrow[v] - mnew);
      float p = __expf(S[v] - mnew);
      float rsum = p;
#pragma unroll
      for (int off = 8; off >= 1; off >>= 1)
        rsum += __shfl_xor(rsum, off, 32);
      lrow[v] = lrow[v] * corr + rsum;
      mrow[v] = mnew;
      o0[v] *= corr;
      o1[v] *= corr;
      sP[v + 8 * half][n] = p;  // C-layout -> LDS for A-layout reload
    }
    // ---- O += P(16x16) @ V_tile(16x32): 4 k-slices x 2 col tiles ----
#pragma unroll
    for (int s = 0; s < 4; ++s) {
      v2f pa = *(const v2f*)&sP[n][4 * s + 2 * half];
      const float* vb = &V[(size_t)(k0 + 4 * s + 2 * half) * DD + h * DKK];
      v2f b0, b1;
      b0.x = vb[n];       b0.y = vb[DD + n];
      b1.x = vb[16 + n];  b1.y = vb[DD + 16 + n];
      o0 = wmma_f32(pa, b0, o0);
      o1 = wmma_f32(pa, b1, o1);
    }
  }
  // ---- normalize and write context [N, D] (col block = head slice) ----
#pragma unroll
  for (int v = 0; v < 8; ++v) {
    float inv = 1.f / lrow[v];
    int row = q0 + v + 8 * half;
    ctx[(size_t)row * DD + h * DKK + n] = o0[v] * inv;
    ctx[(size_t)row * DD + h * DKK + 16 + n] = o1[v] * inv;
  }
}

// ---------------------------------------------------------------------------
extern "C" void kernel_launch(void* const* d_in, const int* in_sizes, int n_in,
                              void* d_out, int out_size, void* d_ws,
                              size_t ws_size, hipStream_t stream) {
  (void)in_sizes; (void)n_in; (void)out_size; (void)ws_size;
  const float* X = (const float*)d_in[0];
  const long long* ei = (const long long*)d_in[1];
  const float* ew = (const float*)d_in[2];
  const float* wq = (const float*)d_in[3];
  const float* bq = (const float*)d_in[4];
  const float* wk = (const float*)d_in[5];
  const float* bk = (const float*)d_in[6];
  const float* wv = (const float*)d_in[7];
  const float* bv = (const float*)d_in[8];
  const float* wo = (const float*)d_in[9];
  const float* bo = (const float*)d_in[10];
  const float* we = (const float*)d_in[11];
  const float* be = (const float*)d_in[12];

  float* ws = (float*)d_ws;
  const size_t NDf = (size_t)NN * DD;   // 1,048,576 floats
  float* Q   = ws;
  float* Kp  = ws + 1 * NDf;
  float* Vp  = ws + 2 * NDf;
  float* ctx = ws + 3 * NDf;
  float* Wd  = ws + 4 * NDf;            // dense NxN edge weights (64 MB)

  dim3 gGemm(NN / 16, DD / 64);
  gemm_xwT_kernel<<<gGemm, 32, 0, stream>>>(X, wq, bq, Q);
  gemm_xwT_kernel<<<gGemm, 32, 0, stream>>>(X, wk, bk, Kp);
  gemm_xwT_kernel<<<gGemm, 32, 0, stream>>>(X, wv, bv, Vp);

  init_dense_kernel<<<((size_t)NN * NN / 4) / 256, 256, 0, stream>>>((float4*)Wd);
  scatter_kernel<<<(EE + 255) / 256, 256, 0, stream>>>(ei, ew, Wd);

  attn_kernel<<<dim3(NN / 16, HH), 32, 0, stream>>>(Q, Kp, Vp, Wd, we, be, ctx);

  gemm_xwT_kernel<<<gGemm, 32, 0, stream>>>(ctx, wo, bo, (float*)d_out);
}